// QuantumEntanglingLinear_new_43215960932472
// MI455X (gfx1250) — compile-verified
//
#include <hip/hip_runtime.h>
#include <hip/hip_bf16.h>

// ---------------------------------------------------------------------------
// QuantumEntanglingLinear on MI455X (gfx1250)
//
// Algebra: out = x @ M^T with M = C*U4 * C*U3 * C*U2 * C*U1 (1024x1024 fp32).
// Each single-qubit gate is R(h) with h = (a0+a1+a2)/2 (2D rotations compose
// by angle addition). C is a permutation (row gather) read off cnot_matrix.
//
// Phase 1 (tiny): build M in workspace via 10 butterfly sweeps/layer + permute.
// Phase 2 (dominant, memory-bound ~11us @ 23.3 TB/s): 32768x1024x1024 fp32
// GEMM on V_WMMA_F32_16X16X4_F32, with double-buffered
// GLOBAL_LOAD_ASYNC_TO_LDS_B128 tile fills overlapped with the matrix pipe
// (ASYNCcnt / s_wait_asynccnt pipelining).
// ---------------------------------------------------------------------------

typedef __attribute__((ext_vector_type(2))) float v2f;
typedef __attribute__((ext_vector_type(8))) float v8f;

#define QDIM 1024
#define NQ   10
#define NLAYERS 4

// ---------------- perm extraction: perm[i] = column of the 1 in row i -------
__global__ void qel_perm_kernel(const float* __restrict__ cnot,
                                int* __restrict__ perm) {
    int i = blockIdx.x * blockDim.x + threadIdx.x;
    if (i < QDIM) {
        int p = 0;
        const float* row = cnot + (size_t)i * QDIM;
        for (int j = 0; j < QDIM; ++j) {
            if (row[j] != 0.0f) p = j;
        }
        perm[i] = p;
    }
}

// ---------------- layer build: M <- C * U_l * M (one column per block) ------
__global__ __launch_bounds__(256)
void qel_layer_kernel(const float* __restrict__ angles,  // [L][NQ][3]
                      const int* __restrict__ perm,
                      float* __restrict__ M,
                      int layer, int first) {
    __shared__ float col[QDIM];
    const int c   = blockIdx.x;
    const int tid = threadIdx.x;

    // load column (identity for layer 0 -> stateless across graph replays)
    for (int i = tid; i < QDIM; i += 256) {
        col[i] = first ? ((i == c) ? 1.0f : 0.0f) : M[(size_t)i * QDIM + c];
    }
    __syncthreads();

    // 10 single-qubit butterfly sweeps. kron order: qubit q lives on bit 9-q.
    for (int b = 0; b < NQ; ++b) {
        const int q = (NQ - 1) - b;
        const float h = 0.5f * (angles[(layer * NQ + q) * 3 + 0] +
                                angles[(layer * NQ + q) * 3 + 1] +
                                angles[(layer * NQ + q) * 3 + 2]);
        const float cs = __cosf(h);
        const float sn = __sinf(h);
        const int bit = 1 << b;
        // 512 disjoint pairs; each handled fully by one thread -> no hazard
        for (int p = tid; p < QDIM / 2; p += 256) {
            const int low  = p & (bit - 1);
            const int high = (p >> b) << (b + 1);
            const int i0 = high | low;
            const int i1 = i0 | bit;
            const float a0 = col[i0];
            const float a1 = col[i1];
            col[i0] = cs * a0 - sn * a1;
            col[i1] = sn * a0 + cs * a1;
        }
        __syncthreads();
    }

    // permuted store: (C * v)[i] = v[perm[i]]
    for (int i = tid; i < QDIM; i += 256) {
        M[(size_t)i * QDIM + c] = col[perm[i]];
    }
}

// ---------------- main GEMM: out(32768x1024) = x @ M^T ----------------------
// Block: 256 threads (8 wave32), tile 128x128, double-buffered K-tiles of 16
// filled by GLOBAL_LOAD_ASYNC_TO_LDS_B128 (ASYNCcnt), overlapped with WMMA.
// Wave grid 2x4; each wave computes 64x32 via 4x2 accumulators of 16x16.
#define BM 128
#define BN 128
#define KT 16
#define NT (QDIM / KT)
#define LDP 20   // padded LDS row stride (floats): 80B = 16B-aligned, no bank conflicts

__device__ __forceinline__ uint32_t lds_off32(const void* p) {
    // LDS aperture keeps the byte offset in addr[31:0] (ISA 10.2 aperture rules)
    return (uint32_t)(uintptr_t)p;
}

__global__ __launch_bounds__(256)
void qel_gemm_wmma_kernel(const float* __restrict__ x,
                          const float* __restrict__ M,
                          float* __restrict__ out) {
    __shared__ __align__(16) float xs[2][BM][LDP];
    __shared__ __align__(16) float ms[2][BN][LDP];

    const int t       = threadIdx.x;
    const int lane    = t & 31;
    const int wid     = t >> 5;          // 0..7
    const int wave_m  = wid >> 2;        // 0..1  -> 64-row slab
    const int wave_n  = wid & 3;         // 0..3  -> 32-col slab
    const int half    = lane >> 4;       // 0/1 (K pair select per ISA layout)
    const int l16     = lane & 15;       // M/N index within fragment

    const int nBase   = blockIdx.x * BN;
    const int rowBase = blockIdx.y * BM;

    v8f acc[4][2];
    #pragma unroll
    for (int i = 0; i < 4; ++i)
        #pragma unroll
        for (int j = 0; j < 2; ++j)
            acc[i][j] = (v8f){0.f,0.f,0.f,0.f,0.f,0.f,0.f,0.f};

    // async DMA fill of one 128x16 x-tile and 128x16 M-tile (4 ops/thread)
    auto issue_stage = [&](int k0, int buf) {
        #pragma unroll
        for (int i = 0; i < 2; ++i) {
            const int f   = t + i * 256;       // float4 slot 0..511
            const int row = f >> 2;            // 0..127
            const int c4  = (f & 3) << 2;      // 0,4,8,12
            const uint32_t xdst = lds_off32(&xs[buf][row][c4]);
            const uint32_t xoff = (uint32_t)(((rowBase + row) * QDIM + k0 + c4) * 4);
            asm volatile("global_load_async_to_lds_b128 %0, %1, %2"
                         :: "v"(xdst), "v"(xoff), "s"(x) : "memory");
            const uint32_t mdst = lds_off32(&ms[buf][row][c4]);
            const uint32_t moff = (uint32_t)(((nBase + row) * QDIM + k0 + c4) * 4);
            asm volatile("global_load_async_to_lds_b128 %0, %1, %2"
                         :: "v"(mdst), "v"(moff), "s"(M) : "memory");
        }
    };

    issue_stage(0, 0);

    for (int s = 0; s < NT; ++s) {
        const int buf = s & 1;
        if (s + 1 < NT) {
            issue_stage((s + 1) * KT, buf ^ 1);
            // async ops retire in order: <=4 in flight => stage s has landed
            asm volatile("s_wait_asynccnt 0x4" ::: "memory");
        } else {
            asm volatile("s_wait_asynccnt 0x0" ::: "memory");
        }
        __syncthreads();

        #pragma unroll
        for (int k4 = 0; k4 < KT; k4 += 4) {
            const int kk = k4 + half * 2;      // per-lane K offset (ISA A/B layout)
            // A fragments: 16x4 of x, lanes 0-15 hold K=0,1; lanes 16-31 K=2,3
            v2f a[4];
            #pragma unroll
            for (int sm = 0; sm < 4; ++sm) {
                const int ar = wave_m * 64 + sm * 16 + l16;
                a[sm].x = xs[buf][ar][kk];
                a[sm].y = xs[buf][ar][kk + 1];
            }
            // B fragments: 4x16 of M^T -> B[k][n] = M[n][k]
            v2f b[2];
            #pragma unroll
            for (int sn = 0; sn < 2; ++sn) {
                const int bc = wave_n * 32 + sn * 16 + l16;
                b[sn].x = ms[buf][bc][kk];
                b[sn].y = ms[buf][bc][kk + 1];
            }
            #pragma unroll
            for (int sm = 0; sm < 4; ++sm)
                #pragma unroll
                for (int sn = 0; sn < 2; ++sn)
                    acc[sm][sn] = __builtin_amdgcn_wmma_f32_16x16x4_f32(
                        /*neg_a=*/false, a[sm], /*neg_b=*/false, b[sn],
                        /*c_mod=*/(short)0, acc[sm][sn],
                        /*reuse_a=*/false, /*reuse_b=*/false);
        }
        __syncthreads();   // all waves done with buf before it is refilled
    }

    // epilogue: C/D layout: VGPR r -> row r (lanes 0-15) / r+8 (lanes 16-31)
    #pragma unroll
    for (int sm = 0; sm < 4; ++sm) {
        #pragma unroll
        for (int sn = 0; sn < 2; ++sn) {
            const int gcol  = nBase + wave_n * 32 + sn * 16 + l16;
            const int rbase = rowBase + wave_m * 64 + sm * 16 + half * 8;
            float* op = out + (size_t)rbase * QDIM + gcol;
            v8f cacc = acc[sm][sn];
            #pragma unroll
            for (int r = 0; r < 8; ++r) {
                op[(size_t)r * QDIM] = cacc[r];
            }
        }
    }
}

// ---------------------------------------------------------------------------
extern "C" void kernel_launch(void* const* d_in, const int* in_sizes, int n_in,
                              void* d_out, int out_size, void* d_ws, size_t ws_size,
                              hipStream_t stream) {
    const float* x      = (const float*)d_in[0];   // [32768][1024]
    const float* angles = (const float*)d_in[1];   // [4][10][3]
    const float* cnot   = (const float*)d_in[2];   // [1024][1024]
    float* out = (float*)d_out;

    float* Mw  = (float*)d_ws;                                  // 4 MB
    int*  perm = (int*)((char*)d_ws + (size_t)QDIM * QDIM * 4); // 4 KB

    // 1) extract CNOT permutation
    qel_perm_kernel<<<dim3((QDIM + 255) / 256), dim3(256), 0, stream>>>(cnot, perm);

    // 2) build combined matrix M = C*U4*C*U3*C*U2*C*U1
    for (int l = 0; l < NLAYERS; ++l) {
        qel_layer_kernel<<<dim3(QDIM), dim3(256), 0, stream>>>(
            angles, perm, Mw, l, l == 0 ? 1 : 0);
    }

    // 3) out = x @ M^T  (memory-bound fp32 WMMA GEMM, async double-buffered)
    dim3 grid(QDIM / BN, 32768 / BM);  // (8, 256)
    qel_gemm_wmma_kernel<<<grid, dim3(256), 0, stream>>>(x, Mw, out);
}